// StaticAdaptiveAdjacencyLayer_40029095199103
// MI455X (gfx1250) — compile-verified
//
#include <hip/hip_runtime.h>
#include <hip/hip_bf16.h>
#include <cstdint>

// Reference: B=32, N=1024.
//   edge_index[0][g] = (g >> 10) & 1023   (row; per-sample wrap is automatic since N*N = 2^20)
//   edge_index[1][g] =  g        & 1023   (col)
//   edge_attr[g]     = sigmoid(V.flat[g])
//
// d_out layout (flat, return order): [row: total int32][col: total int32][attr: total f32]
//
// Pure streaming kernel: 512 MB total traffic -> ~22 us floor at 23.3 TB/s.
// 128-bit VMEM ops; non-temporal stores (output is write-once, 384 MB > L2),
// regular-temporal loads (input 128 MB fits in 192 MB L2, re-read per replay).

typedef __attribute__((ext_vector_type(4))) float float4v;
typedef __attribute__((ext_vector_type(4))) int   int4v;

__device__ __forceinline__ float fast_sigmoid(float x) {
  // sigmoid(x) = 1/(1+exp(-x)); |x| ~ N(0,1) so no overflow concerns.
  return 1.0f / (1.0f + __expf(-x));
}

__global__ __launch_bounds__(256) void adj_vec4_kernel(
    const float* __restrict__ V,
    int*   __restrict__ out_row,
    int*   __restrict__ out_col,
    float* __restrict__ out_attr,
    long long n_vec4) {
  long long v = (long long)blockIdx.x * blockDim.x + threadIdx.x;
  if (v >= n_vec4) return;
  long long g = v << 2;  // first element of this thread's group of 4

  // Row is constant across 4 consecutive elements (row changes every 1024).
  int row  = (int)((g >> 10) & 1023);
  int col0 = (int)( g        & 1023);
  int4v rows = { row, row, row, row };
  int4v cols = { col0, col0 + 1, col0 + 2, col0 + 3 };

  // 16B load (regular temporal: keep input resident in L2 across replays).
  const float4v* Vv = (const float4v*)V;
  float4v x = Vv[v];

  float4v s;
  s.x = fast_sigmoid(x.x);
  s.y = fast_sigmoid(x.y);
  s.z = fast_sigmoid(x.z);
  s.w = fast_sigmoid(x.w);

  // 3x 16B non-temporal stores (global_store_b128 th:NT): output is never
  // re-read; don't let 384 MB of stores thrash the 192 MB L2.
  __builtin_nontemporal_store(rows, (int4v*)  (out_row ) + v);
  __builtin_nontemporal_store(cols, (int4v*)  (out_col ) + v);
  __builtin_nontemporal_store(s,    (float4v*)(out_attr) + v);
}

// Scalar tail (not hit for the reference shape 2^25, kept for robustness).
__global__ __launch_bounds__(64) void adj_tail_kernel(
    const float* __restrict__ V,
    int*   __restrict__ out_row,
    int*   __restrict__ out_col,
    float* __restrict__ out_attr,
    long long start, long long total) {
  long long g = start + (long long)blockIdx.x * blockDim.x + threadIdx.x;
  if (g >= total) return;
  out_row[g]  = (int)((g >> 10) & 1023);
  out_col[g]  = (int)( g        & 1023);
  out_attr[g] = fast_sigmoid(V[g]);
}

extern "C" void kernel_launch(void* const* d_in, const int* in_sizes, int n_in,
                              void* d_out, int out_size, void* d_ws, size_t ws_size,
                              hipStream_t stream) {
  const float* V = (const float*)d_in[0];
  const long long total = (long long)in_sizes[0];  // B*N*N = 33,554,432

  int*   out_row  = (int*)d_out;
  int*   out_col  = out_row + total;
  float* out_attr = (float*)(out_col + total);

  const long long n_vec4 = total >> 2;
  const int threads = 256;
  const long long blocks = (n_vec4 + threads - 1) / threads;
  adj_vec4_kernel<<<(unsigned)blocks, threads, 0, stream>>>(
      V, out_row, out_col, out_attr, n_vec4);

  const long long rem_start = n_vec4 << 2;
  const long long rem = total - rem_start;
  if (rem > 0) {
    adj_tail_kernel<<<(unsigned)((rem + 63) / 64), 64, 0, stream>>>(
        V, out_row, out_col, out_attr, rem_start, total);
  }
}